// EQGATEdgeVirtualGNN_77369540870667
// MI455X (gfx1250) — compile-verified
//
#include <hip/hip_runtime.h>
#include <math.h>

// ---------------- problem constants (from reference) ----------------
#define SDIM 128
#define VDIM 32
#define EDIM 32
#define NLAYERS 5
#define BSG 48
#define MPG 40                       // nodes per graph
#define NNODES (BSG * MPG)           // 1920
#define EPG (MPG * (MPG - 1))        // 1560 edges per graph
#define NEDGES (BSG * EPG)           // 74880
#define EPSF 1e-6f
#define EN2_NPAD 256                 // padded N for edge_net2 weights

typedef float v2f __attribute__((ext_vector_type(2)));
typedef float v8f __attribute__((ext_vector_type(8)));

// FP32 WMMA: D(16x16) = A(16x4) * B(4x16) + C   -> v_wmma_f32_16x16x4_f32
__device__ __forceinline__ v8f wmma4(v2f a, v2f b, v8f c) {
  return __builtin_amdgcn_wmma_f32_16x16x4_f32(false, a, false, b, (short)0, c,
                                               false, false);
}

__device__ __forceinline__ float silu_f(float x) {
  return x / (1.f + __expf(-x));
}

// ---------------- generic WMMA GEMM -------------------------------------
// C = act(A@W + bias) [+= C].  A: M x KDIM row-major (lda), W: KDIM x ldw
// row-major, C: M x Nreal (ldc).  One wave computes a 16 x (16*TNSUB) tile.
// Requirements: M % 16 == 0, KDIM % 4 == 0, ldw % (16*TNSUB) == 0.
// NG: guard epilogue stores with nn < Nreal (loads always in-bounds via ldw).
template <int KDIM, int TNSUB, bool NG>
__global__ void gemm_t(const float* __restrict__ A, int lda,
                       const float* __restrict__ W, int ldw,
                       const float* __restrict__ bias, float* __restrict__ C,
                       int ldc, int M, int Nreal, int act, int accum) {
  int w = blockIdx.x * (blockDim.x >> 5) + (threadIdx.x >> 5);
  int lane = threadIdx.x & 31;
  int tilesN = ldw / (16 * TNSUB);
  int tilesM = M >> 4;
  if (w >= tilesM * tilesN) return;
  int tm = w / tilesN, tn = w % tilesN;
  int m = tm * 16 + (lane & 15);
  int nbase = tn * (16 * TNSUB) + (lane & 15);
  int khalf = (lane >> 4) << 1;  // lanes 0-15 -> k+0/1, lanes 16-31 -> k+2/3
  const float* Arow = A + (size_t)m * lda + khalf;
  const float* Wcol = W + (size_t)khalf * ldw + nbase;
  v8f zero = {};
  v8f acc[TNSUB];
#pragma unroll
  for (int t = 0; t < TNSUB; ++t) acc[t] = zero;
#pragma unroll 4
  for (int k0 = 0; k0 < KDIM; k0 += 4) {
    v2f av;
    av.x = Arow[k0];
    av.y = Arow[k0 + 1];
    const float* wr0 = Wcol + (size_t)k0 * ldw;
#pragma unroll
    for (int t = 0; t < TNSUB; ++t) {
      v2f bv;
      bv.x = wr0[t * 16];
      bv.y = wr0[ldw + t * 16];
      acc[t] = wmma4(av, bv, acc[t]);
    }
  }
  int mrow0 = tm * 16 + ((lane >> 4) << 3);
#pragma unroll
  for (int t = 0; t < TNSUB; ++t) {
    int nn = nbase + t * 16;
    if (!NG || nn < Nreal) {
      float bval = bias ? bias[nn] : 0.f;
#pragma unroll
      for (int r = 0; r < 8; ++r) {
        size_t off = (size_t)(mrow0 + r) * ldc + nn;
        float val = acc[t][r] + bval;
        if (act) val = silu_f(val);
        if (accum) val += C[off];
        C[off] = val;
      }
    }
  }
}

// -------- edge_net1: h1 = silu([s[dst], s[src], de, a, d] @ W + b) -------
// K = 290, N = 128, M = NEDGES.  Wave tile: 16 x 64 (TNSUB = 4).
// Concat handled as wave-uniform segment loops; the 2-element scalar tail
// (a, d) is a single fixed step with zeros in lanes 16-31 (k = 290, 291),
// whose B rows are clamped in-bounds (product is zero regardless).
__global__ void edge_net1_kernel(const float* __restrict__ s,
                                 const float* __restrict__ de,
                                 const float* __restrict__ aA,
                                 const float* __restrict__ dA,
                                 const int* __restrict__ srcI,
                                 const int* __restrict__ dstI,
                                 const float* __restrict__ W,
                                 const float* __restrict__ bias,
                                 float* __restrict__ h1) {
  const int TNS = 4;
  int w = blockIdx.x * (blockDim.x >> 5) + (threadIdx.x >> 5);
  int lane = threadIdx.x & 31;
  const int tilesN = SDIM / (16 * TNS);  // 2
  const int tilesM = NEDGES / 16;
  if (w >= tilesM * tilesN) return;
  int tm = w / tilesN, tn = w % tilesN;
  int row = tm * 16 + (lane & 15);
  int khalf = (lane >> 4) << 1;
  const float* drow = s + (size_t)dstI[row] * SDIM + khalf;
  const float* srow = s + (size_t)srcI[row] * SDIM + khalf;
  const float* derow = de + (size_t)row * EDIM + khalf;
  float aval = aA[row], dval = dA[row];
  int nbase = tn * (16 * TNS) + (lane & 15);
  const float* Wc = W + (size_t)khalf * SDIM + nbase;
  v8f zero = {};
  v8f acc[TNS];
#pragma unroll
  for (int t = 0; t < TNS; ++t) acc[t] = zero;

  // segment 1: k in [0,128)  <- s[dst]
#pragma unroll 4
  for (int k0 = 0; k0 < SDIM; k0 += 4) {
    v2f av;
    av.x = drow[k0];
    av.y = drow[k0 + 1];
    const float* wr0 = Wc + (size_t)k0 * SDIM;
#pragma unroll
    for (int t = 0; t < TNS; ++t) {
      v2f bv;
      bv.x = wr0[t * 16];
      bv.y = wr0[SDIM + t * 16];
      acc[t] = wmma4(av, bv, acc[t]);
    }
  }
  // segment 2: k in [128,256)  <- s[src]
#pragma unroll 4
  for (int k0 = 0; k0 < SDIM; k0 += 4) {
    v2f av;
    av.x = srow[k0];
    av.y = srow[k0 + 1];
    const float* wr0 = Wc + (size_t)(SDIM + k0) * SDIM;
#pragma unroll
    for (int t = 0; t < TNS; ++t) {
      v2f bv;
      bv.x = wr0[t * 16];
      bv.y = wr0[SDIM + t * 16];
      acc[t] = wmma4(av, bv, acc[t]);
    }
  }
  // segment 3: k in [256,288)  <- de
#pragma unroll
  for (int k0 = 0; k0 < EDIM; k0 += 4) {
    v2f av;
    av.x = derow[k0];
    av.y = derow[k0 + 1];
    const float* wr0 = Wc + (size_t)(2 * SDIM + k0) * SDIM;
#pragma unroll
    for (int t = 0; t < TNS; ++t) {
      v2f bv;
      bv.x = wr0[t * 16];
      bv.y = wr0[SDIM + t * 16];
      acc[t] = wmma4(av, bv, acc[t]);
    }
  }
  // tail: k = 288 (a), 289 (d); lanes 16-31 carry k = 290/291 -> zero A,
  // B rows clamped to 288/289 (in-bounds; multiplied by zero).
  {
    bool lo = (lane < 16);
    v2f av;
    av.x = lo ? aval : 0.f;
    av.y = lo ? dval : 0.f;
    const float* wr0 = W + (size_t)(2 * SDIM + EDIM) * SDIM + nbase;
#pragma unroll
    for (int t = 0; t < TNS; ++t) {
      v2f bv;
      bv.x = wr0[t * 16];
      bv.y = wr0[SDIM + t * 16];
      acc[t] = wmma4(av, bv, acc[t]);
    }
  }
  int mrow0 = tm * 16 + ((lane >> 4) << 3);
#pragma unroll
  for (int t = 0; t < TNS; ++t) {
    int nn = nbase + t * 16;
    float bval = bias[nn];
#pragma unroll
    for (int r = 0; r < 8; ++r) {
      h1[(size_t)(mrow0 + r) * SDIM + nn] = silu_f(acc[t][r] + bval);
    }
  }
}

// -------- concat-2 GEMM: C = act([A0(128) | A1(32)] @ W + b) -------------
// K = 160, two wave-uniform segments.  N (=ldw) % (16*TNSUB) == 0.
template <int TNSUB>
__global__ void gemm_cat_t(const float* __restrict__ A0,
                           const float* __restrict__ A1,
                           const float* __restrict__ W, int ldw,
                           const float* __restrict__ bias,
                           float* __restrict__ C, int M, int act) {
  int w = blockIdx.x * (blockDim.x >> 5) + (threadIdx.x >> 5);
  int lane = threadIdx.x & 31;
  int tilesN = ldw / (16 * TNSUB);
  int tilesM = M >> 4;
  if (w >= tilesM * tilesN) return;
  int tm = w / tilesN, tn = w % tilesN;
  int m = tm * 16 + (lane & 15);
  int khalf = (lane >> 4) << 1;
  const float* a0row = A0 + (size_t)m * SDIM + khalf;
  const float* a1row = A1 + (size_t)m * VDIM + khalf;
  int nbase = tn * (16 * TNSUB) + (lane & 15);
  const float* Wc = W + (size_t)khalf * ldw + nbase;
  v8f zero = {};
  v8f acc[TNSUB];
#pragma unroll
  for (int t = 0; t < TNSUB; ++t) acc[t] = zero;
#pragma unroll 4
  for (int k0 = 0; k0 < SDIM; k0 += 4) {
    v2f av;
    av.x = a0row[k0];
    av.y = a0row[k0 + 1];
    const float* wr0 = Wc + (size_t)k0 * ldw;
#pragma unroll
    for (int t = 0; t < TNSUB; ++t) {
      v2f bv;
      bv.x = wr0[t * 16];
      bv.y = wr0[ldw + t * 16];
      acc[t] = wmma4(av, bv, acc[t]);
    }
  }
#pragma unroll
  for (int k0 = 0; k0 < VDIM; k0 += 4) {
    v2f av;
    av.x = a1row[k0];
    av.y = a1row[k0 + 1];
    const float* wr0 = Wc + (size_t)(SDIM + k0) * ldw;
#pragma unroll
    for (int t = 0; t < TNSUB; ++t) {
      v2f bv;
      bv.x = wr0[t * 16];
      bv.y = wr0[ldw + t * 16];
      acc[t] = wmma4(av, bv, acc[t]);
    }
  }
  int mrow0 = tm * 16 + ((lane >> 4) << 3);
#pragma unroll
  for (int t = 0; t < TNSUB; ++t) {
    int nn = nbase + t * 16;
    float bval = bias[nn];
#pragma unroll
    for (int r = 0; r < 8; ++r) {
      float val = acc[t][r] + bval;
      if (act) val = silu_f(val);
      C[(size_t)(mrow0 + r) * ldw + nn] = val;
    }
  }
}

// -------- pad edge_net2 weights K x Nreal -> K x Npad (zero fill) --------
__global__ void pad_w_kernel(float* __restrict__ dst,
                             const float* __restrict__ src, int K, int Nreal,
                             int Npad) {
  int idx = blockIdx.x * blockDim.x + threadIdx.x;
  if (idx >= K * Npad) return;
  int k = idx / Npad, n = idx % Npad;
  dst[idx] = (n < Nreal) ? src[k * Nreal + n] : 0.f;
}

// -------- layernorm stats: s += g (broadcast), per-graph mu/var/vn -------
__global__ void ln_stats_kernel(float* __restrict__ s,
                                const float* __restrict__ v,
                                const float* __restrict__ g48,
                                float* __restrict__ stats) {
  int b = blockIdx.x, tid = threadIdx.x;
  float s1 = 0.f, s2 = 0.f, v2 = 0.f;
  for (int i = tid; i < MPG * SDIM; i += blockDim.x) {
    int node = b * MPG + i / SDIM, c = i % SDIM;
    float val = s[node * SDIM + c] + g48[b * SDIM + c];
    s[node * SDIM + c] = val;
    s1 += val;
    s2 += val * val;
  }
  for (int i = tid; i < MPG * 3 * VDIM; i += blockDim.x) {
    float x = v[b * MPG * 3 * VDIM + i];
    v2 += x * x;
  }
  __shared__ float r1[256], r2[256], r3[256];
  r1[tid] = s1; r2[tid] = s2; r3[tid] = v2;
  __syncthreads();
  for (int st = 128; st > 0; st >>= 1) {
    if (tid < st) {
      r1[tid] += r1[tid + st];
      r2[tid] += r2[tid + st];
      r3[tid] += r3[tid + st];
    }
    __syncthreads();
  }
  if (tid == 0) {
    float mu = r1[0] / (float)(MPG * SDIM);
    float var = r2[0] / (float)(MPG * SDIM) - mu * mu;
    float vn = r3[0] / (float)(MPG * VDIM);
    stats[b * 3 + 0] = mu;
    stats[b * 3 + 1] = var;
    stats[b * 3 + 2] = vn;
  }
}

__global__ void ln_apply_kernel(float* __restrict__ s, float* __restrict__ v,
                                const float* __restrict__ stats,
                                const float* __restrict__ lng,
                                const float* __restrict__ lnb) {
  int idx = blockIdx.x * blockDim.x + threadIdx.x;
  if (idx < NNODES * SDIM) {
    int b = idx / (MPG * SDIM);
    int c = idx % SDIM;
    float mu = stats[b * 3], var = stats[b * 3 + 1];
    s[idx] = (s[idx] - mu) * rsqrtf(fmaxf(var, EPSF)) * lng[c] + lnb[c];
  }
  if (idx < NNODES * 3 * VDIM) {
    int b = idx / (MPG * 3 * VDIM);
    v[idx] *= rsqrtf(fmaxf(stats[b * 3 + 2], EPSF));
  }
}

// -------- deterministic per-dst aggregation (fully connected graphs) -----
__global__ void aggregate_kernel(const float* __restrict__ h2, int outw,
                                 const float* __restrict__ sn,
                                 const float* __restrict__ vb,  // null at L0
                                 const float* __restrict__ rn,
                                 float* __restrict__ s, float* __restrict__ v,
                                 float* __restrict__ p) {
  int n = blockIdx.x;     // node
  int tid = threadIdx.x;  // 128 threads
  int b = n / MPG, j = n % MPG;
  int t = tid >> 5, cc = tid & 31;
  float accS = 0.f, accV = 0.f, accP = 0.f;
  const int base = SDIM + 1 + EDIM;  // 161
  for (int ai = 0; ai < MPG; ++ai) {
    if (ai == j) continue;
    int e = b * EPG + ai * (MPG - 1) + (j < ai ? j : j - 1);
    const float* hrow = h2 + (size_t)e * outw;
    int srcN = b * MPG + ai;
    accS += hrow[tid] * sn[srcN * SDIM + tid];
    if (tid < 96) {
      float contrib = rn[e * 3 + t] * hrow[base + cc];
      if (vb) contrib += hrow[base + VDIM + cc] * vb[srcN * 96 + t * VDIM + cc];
      accV += contrib;
    }
    if (tid < 3) accP += hrow[SDIM] * rn[e * 3 + tid];
  }
  const float inv_deg = 1.f / (float)(MPG - 1);
  s[n * SDIM + tid] += accS;
  if (tid < 96) v[n * 96 + tid] += accV * inv_deg;
  if (tid < 3) p[n * 3 + tid] += accP * inv_deg;
}

// -------- gated equivariant pieces --------------------------------------
__global__ void vnorm_kernel(const float* __restrict__ vv,
                             float* __restrict__ vno) {
  int idx = blockIdx.x * blockDim.x + threadIdx.x;
  if (idx >= NNODES * VDIM) return;
  int n = idx / VDIM, c = idx % VDIM;
  float acc = 0.f;
#pragma unroll
  for (int t = 0; t < 3; ++t) {
    float x = vv[(n * 3 + t) * 64 + c];
    acc += x * x;
  }
  vno[idx] = sqrtf(fmaxf(acc, EPSF));
}

__global__ void apply_gate_kernel(const float* __restrict__ hout,
                                  const float* __restrict__ vv,
                                  float* __restrict__ s,
                                  float* __restrict__ vtmp) {
  int n = blockIdx.x;
  int tid = threadIdx.x;  // 128
  s[n * SDIM + tid] += hout[n * 160 + VDIM + tid];
  if (tid < 96) {
    int t = tid >> 5, c = tid & 31;
    vtmp[n * 96 + tid] = hout[n * 160 + c] * vv[(n * 3 + t) * 64 + VDIM + c];
  }
}

__global__ void vadd_kernel(float* __restrict__ v,
                            const float* __restrict__ vtmp) {
  int idx = blockIdx.x * blockDim.x + threadIdx.x;
  if (idx < NNODES * 3 * VDIM) v[idx] += vtmp[idx];
}

// -------- geometry ------------------------------------------------------
__global__ void p_center_kernel(float* __restrict__ p) {
  int b = blockIdx.x, tid = threadIdx.x;  // 128
  __shared__ float mean[3];
  if (tid < 3) {
    float acc = 0.f;
    for (int j = 0; j < MPG; ++j) acc += p[(b * MPG + j) * 3 + tid];
    mean[tid] = acc / (float)MPG;
  }
  __syncthreads();
  if (tid < MPG * 3) p[b * MPG * 3 + tid] -= mean[tid % 3];
}

__global__ void edge_geom_kernel(const float* __restrict__ p,
                                 const int* __restrict__ srcI,
                                 const int* __restrict__ dstI,
                                 float* __restrict__ dE, float* __restrict__ aE,
                                 float* __restrict__ rn) {
  int e = blockIdx.x * blockDim.x + threadIdx.x;
  if (e >= NEDGES) return;
  int si = srcI[e], di = dstI[e];
  float r0 = p[di * 3 + 0] - p[si * 3 + 0];
  float r1 = p[di * 3 + 1] - p[si * 3 + 1];
  float r2 = p[di * 3 + 2] - p[si * 3 + 2];
  float a = p[di * 3 + 0] * p[si * 3 + 0] + p[di * 3 + 1] * p[si * 3 + 1] +
            p[di * 3 + 2] * p[si * 3 + 2];
  float dd = sqrtf(fmaxf(r0 * r0 + r1 * r1 + r2 * r2, EPSF));
  float inv = 1.f / dd;
  dE[e] = dd;
  aE[e] = a;
  rn[e * 3 + 0] = r0 * inv;
  rn[e * 3 + 1] = r1 * inv;
  rn[e * 3 + 2] = r2 * inv;
}

// -------- virtual node --------------------------------------------------
__global__ void vreduce_kernel(const float* __restrict__ g1,
                               float* __restrict__ gA) {
  int b = blockIdx.x, c = threadIdx.x;  // 128
  float acc = 0.f;
  for (int j = 0; j < MPG; ++j) acc += g1[(b * MPG + j) * SDIM + c];
  gA[b * SDIM + c] = acc;
}

__global__ void final_add_kernel(float* __restrict__ s,
                                 const float* __restrict__ g48) {
  int idx = blockIdx.x * blockDim.x + threadIdx.x;
  if (idx >= NNODES * SDIM) return;
  int b = idx / (MPG * SDIM);
  s[idx] += g48[b * SDIM + idx % SDIM];
}

// ======================= host side ======================================
struct LayerParams {
  const float *en1W, *en1b, *en2W, *en2b, *epostW, *epostb, *epreW, *epreb;
  const float *lnb, *lng, *scW, *scb, *ws1W, *ws1b, *ws2W, *ws2b;
  const float *wv0W, *wv1W, *vecW, *vlinW, *vlinb, *vm1W, *vm1b, *vm2W, *vm2b;
  int outw;
  bool hasV, useMlp;
};

template <int KDIM, int TNSUB, bool NG>
static inline void launch_g(hipStream_t st, const float* A, int lda,
                            const float* W, int ldw, const float* bias,
                            float* C, int ldc, int M, int Nreal, int act,
                            int accum) {
  int tiles = (M / 16) * (ldw / (16 * TNSUB));
  int blocks = (tiles + 7) / 8;
  gemm_t<KDIM, TNSUB, NG><<<blocks, 256, 0, st>>>(A, lda, W, ldw, bias, C, ldc,
                                                  M, Nreal, act, accum);
}

extern "C" void kernel_launch(void* const* d_in, const int* in_sizes, int n_in,
                              void* d_out, int out_size, void* d_ws,
                              size_t ws_size, hipStream_t stream) {
  (void)in_sizes; (void)n_in; (void)out_size; (void)ws_size;
  auto F = [&](int i) { return (const float*)d_in[i]; };

  // --- unflatten params: top-level dict order, nested jax (sorted keys) ---
  LayerParams L[NLAYERS];
  int base = 7;
  for (int i = 0; i < NLAYERS; ++i) {
    bool hasV = (i > 0), useMlp = (i < NLAYERS - 1);
    LayerParams& P = L[i];
    int k = base;
    P.en1W = F(k++);  P.en1b = F(k++);
    P.en2W = F(k++);  P.en2b = F(k++);
    P.epostW = F(k++); P.epostb = F(k++);
    P.epreW = F(k++);  P.epreb = F(k++);
    P.lnb = F(k++);    P.lng = F(k++);
    P.scW = F(k++);    P.scb = F(k++);
    P.ws1W = F(k++);   P.ws1b = F(k++);
    if (useMlp) { P.ws2W = F(k++); P.ws2b = F(k++); } else { P.ws2W = nullptr; P.ws2b = nullptr; }
    P.wv0W = F(k++);
    P.wv1W = useMlp ? F(k++) : nullptr;
    P.vecW = hasV ? F(k++) : nullptr;
    P.vlinW = F(k++);  P.vlinb = F(k++);
    P.vm1W = F(k++);   P.vm1b = F(k++);
    P.vm2W = F(k++);   P.vm2b = F(k++);
    P.outw = (hasV ? 2 : 1) * VDIM + SDIM + 1 + EDIM;  // 193 or 225
    P.hasV = hasV; P.useMlp = useMlp;
    base = k;
  }
  const int* ei = (const int*)d_in[base + 1];  // edge_index_global (2 x E)
  const int* eiSrc = ei;
  const int* eiDst = ei + NEDGES;

  // --- output state regions (s, v, e, p concatenated) ---
  float* out = (float*)d_out;
  float* s = out;
  float* v = s + (size_t)NNODES * SDIM;
  float* e = v + (size_t)NNODES * 3 * VDIM;
  float* p = e + (size_t)NEDGES * EDIM;

  // --- workspace carve-out ---
  float* w = (float*)d_ws;
  size_t off = 0;
  auto alloc = [&](size_t n) { float* q = w + off; off += n; return q; };
  float* dE   = alloc(NEDGES);
  float* aE   = alloc(NEDGES);
  float* rn   = alloc((size_t)NEDGES * 3);
  float* de   = alloc((size_t)NEDGES * EDIM);
  float* h1   = alloc((size_t)NEDGES * SDIM);
  float* h2   = alloc((size_t)NEDGES * 225);
  float* sn   = alloc((size_t)NNODES * SDIM);
  float* vb   = alloc((size_t)NNODES * 3 * VDIM);
  float* vv   = alloc((size_t)NNODES * 3 * 64);
  float* vno  = alloc((size_t)NNODES * VDIM);
  float* hmid = alloc((size_t)NNODES * SDIM);
  float* hout = alloc((size_t)NNODES * 160);
  float* vtmp = alloc((size_t)NNODES * 3 * VDIM);
  float* g1   = alloc((size_t)NNODES * SDIM);
  float* gA   = alloc((size_t)BSG * SDIM);
  float* gB   = alloc((size_t)BSG * SDIM);
  float* g48  = alloc((size_t)BSG * SDIM);
  float* stats = alloc((size_t)BSG * 3);
  float* Wpad = alloc((size_t)SDIM * EN2_NPAD);  // padded edge_net2 weights

  // --- init state from inputs (deterministic every call) ---
  hipMemcpyAsync(s, d_in[0], sizeof(float) * NNODES * SDIM,     hipMemcpyDeviceToDevice, stream);
  hipMemcpyAsync(v, d_in[1], sizeof(float) * NNODES * 3 * VDIM, hipMemcpyDeviceToDevice, stream);
  hipMemcpyAsync(p, d_in[2], sizeof(float) * NNODES * 3,        hipMemcpyDeviceToDevice, stream);
  hipMemcpyAsync(dE, d_in[3], sizeof(float) * NEDGES,           hipMemcpyDeviceToDevice, stream);
  hipMemcpyAsync(aE, d_in[4], sizeof(float) * NEDGES,           hipMemcpyDeviceToDevice, stream);
  hipMemcpyAsync(rn, d_in[5], sizeof(float) * NEDGES * 3,       hipMemcpyDeviceToDevice, stream);
  hipMemcpyAsync(e, d_in[6], sizeof(float) * NEDGES * EDIM,     hipMemcpyDeviceToDevice, stream);
  hipMemsetAsync(g48, 0, sizeof(float) * BSG * SDIM, stream);

  for (int li = 0; li < NLAYERS; ++li) {
    const LayerParams& P = L[li];
    const int outw = P.outw;

    // s += g ; layernorm
    ln_stats_kernel<<<BSG, 256, 0, stream>>>(s, v, g48, stats);
    ln_apply_kernel<<<(NNODES * SDIM + 255) / 256, 256, 0, stream>>>(
        s, v, stats, P.lng, P.lnb);

    // de = edge_pre(e)
    launch_g<EDIM, 2, false>(stream, e, EDIM, P.epreW, EDIM, P.epreb, de,
                             EDIM, NEDGES, EDIM, 0, 0);

    // h1 = silu(edge_net1([s[dst], s[src], de, a, d]))  (fused gather)
    {
      int tiles = (NEDGES / 16) * (SDIM / 64);
      edge_net1_kernel<<<(tiles + 7) / 8, 256, 0, stream>>>(
          s, de, aE, dE, eiSrc, eiDst, P.en1W, P.en1b, h1);
    }

    // h2 = edge_net2(h1)   (weights padded to N=256, store-guarded)
    pad_w_kernel<<<(SDIM * EN2_NPAD + 255) / 256, 256, 0, stream>>>(
        Wpad, P.en2W, SDIM, outw, EN2_NPAD);
    launch_g<SDIM, 4, true>(stream, h1, SDIM, Wpad, EN2_NPAD, P.en2b, h2,
                            outw, NEDGES, outw, 0, 0);

    // e += edge_post(gij)   (gij = h2[:, 129:161])
    launch_g<EDIM, 2, false>(stream, h2 + SDIM + 1, outw, P.epostW, EDIM,
                             P.epostb, e, EDIM, NEDGES, EDIM, 0, 1);

    // sn = scalar_net(s);  vb = v @ vector_net (if present)
    launch_g<SDIM, 4, false>(stream, s, SDIM, P.scW, SDIM, P.scb, sn, SDIM,
                             NNODES, SDIM, 0, 0);
    if (P.hasV)
      launch_g<VDIM, 2, false>(stream, v, VDIM, P.vecW, VDIM, nullptr, vb,
                               VDIM, NNODES * 3, VDIM, 0, 0);

    // per-dst gather aggregation -> updates s, v, p
    aggregate_kernel<<<NNODES, 128, 0, stream>>>(
        h2, outw, sn, P.hasV ? vb : nullptr, rn, s, v, p);

    // gated equivariant update
    launch_g<VDIM, 4, false>(stream, v, VDIM, P.wv0W, 2 * VDIM, nullptr, vv,
                             2 * VDIM, NNODES * 3, 2 * VDIM, 0, 0);
    vnorm_kernel<<<(NNODES * VDIM + 255) / 256, 256, 0, stream>>>(vv, vno);
    if (P.useMlp) {
      int tiles1 = (NNODES / 16) * (SDIM / 64);
      gemm_cat_t<4><<<(tiles1 + 7) / 8, 256, 0, stream>>>(
          s, vno, P.ws1W, SDIM, P.ws1b, hmid, NNODES, 1);
      launch_g<SDIM, 2, false>(stream, hmid, SDIM, P.ws2W, VDIM + SDIM,
                               P.ws2b, hout, VDIM + SDIM, NNODES, VDIM + SDIM,
                               0, 0);
    } else {
      int tiles1 = (NNODES / 16) * ((VDIM + SDIM) / 32);
      gemm_cat_t<2><<<(tiles1 + 7) / 8, 256, 0, stream>>>(
          s, vno, P.ws1W, VDIM + SDIM, P.ws1b, hout, NNODES, 0);
    }
    apply_gate_kernel<<<NNODES, 128, 0, stream>>>(hout, vv, s, vtmp);
    if (P.useMlp) {
      // v += vtmp @ upd_Wv1
      launch_g<VDIM, 2, false>(stream, vtmp, VDIM, P.wv1W, VDIM, nullptr, v,
                               VDIM, NNODES * 3, VDIM, 0, 1);
    } else {
      vadd_kernel<<<(NNODES * 3 * VDIM + 255) / 256, 256, 0, stream>>>(v, vtmp);
    }

    // p centering and geometry recompute
    p_center_kernel<<<BSG, 128, 0, stream>>>(p);
    edge_geom_kernel<<<(NEDGES + 255) / 256, 256, 0, stream>>>(p, eiSrc,
                                                               eiDst, dE, aE,
                                                               rn);

    // virtual node: g48 = vmlp2(silu(vmlp1(segsum(vlin(s)))))
    launch_g<SDIM, 4, false>(stream, s, SDIM, P.vlinW, SDIM, P.vlinb, g1,
                             SDIM, NNODES, SDIM, 0, 0);
    vreduce_kernel<<<BSG, SDIM, 0, stream>>>(g1, gA);
    launch_g<SDIM, 4, false>(stream, gA, SDIM, P.vm1W, SDIM, P.vm1b, gB, SDIM,
                             BSG, SDIM, 1, 0);
    launch_g<SDIM, 4, false>(stream, gB, SDIM, P.vm2W, SDIM, P.vm2b, g48,
                             SDIM, BSG, SDIM, 0, 0);
  }

  // final s += g
  final_add_kernel<<<(NNODES * SDIM + 255) / 256, 256, 0, stream>>>(s, g48);
}